// SparseArch_12927851561547
// MI455X (gfx1250) — compile-verified
//
#include <hip/hip_runtime.h>
#include <stdint.h>
#include <stddef.h>

// Problem constants (match reference)
#define F 26
#define D 128
#define NROWS 100000
#define B 4096
#define L 20
#define T (B * L)

#define WAVES_PER_BLOCK 8
#define PIPE 8   // rows in flight per wave (power of two)

typedef float float4v __attribute__((ext_vector_type(4)));
typedef int v4i __attribute__((vector_size(16)));

typedef __attribute__((address_space(1))) v4i gas_v4i;   // global (prints as __device__)
typedef __attribute__((address_space(3))) v4i las_v4i;   // LDS

// ---- CDNA5 async global->LDS load (ASYNCcnt-tracked) --------------------
#if defined(__HIP_DEVICE_COMPILE__) && __has_builtin(__builtin_amdgcn_global_load_async_to_lds_b128)
#define ASYNC_LOAD_B128(gptr, lptr)                                         \
    __builtin_amdgcn_global_load_async_to_lds_b128(                         \
        (gas_v4i*)(gptr), (las_v4i*)(lptr), 0, 0)
#else
#define ASYNC_LOAD_B128(gptr, lptr)                                         \
    asm volatile("global_load_async_to_lds_b128 %0, %1, off"                \
                 :: "v"((uint32_t)(uintptr_t)(lptr)),                       \
                    "v"((uint64_t)(uintptr_t)(gptr))                        \
                 : "memory")
#endif

#if defined(__HIP_DEVICE_COMPILE__) && __has_builtin(__builtin_amdgcn_s_wait_asynccnt)
#define WAIT_ASYNC(n) __builtin_amdgcn_s_wait_asynccnt(n)
#else
#define WAIT_ASYNC(n) asm volatile("s_wait_asynccnt " #n ::: "memory")
#endif

#if defined(__HIP_DEVICE_COMPILE__) && __has_builtin(__builtin_amdgcn_s_wait_dscnt)
#define WAIT_DS0() __builtin_amdgcn_s_wait_dscnt(0)
#else
#define WAIT_DS0() asm volatile("s_wait_dscnt 0" ::: "memory")
#endif

// One wave32 per (feature, bag). Each lane owns 4 consecutive dims -> one
// 512B global_load_async_to_lds_b128 per embedding row per wave.
__global__ __launch_bounds__(WAVES_PER_BLOCK * 32)
void embag_pool_kernel(const float* __restrict__ tables,        // [F, N, D]
                       const long long* __restrict__ values,    // [F, T]
                       const long long* __restrict__ offsets,   // [F, B+1]
                       float* __restrict__ out)                 // [B, F, D]
{
    __shared__ float stage[WAVES_PER_BLOCK][PIPE][D];   // 32 KB / block

    const int lane = threadIdx.x & 31;
    const int wv   = threadIdx.x >> 5;
    const int g    = blockIdx.x * WAVES_PER_BLOCK + wv; // bag id in [0, F*B)
    const int f    = g / B;
    const int b    = g - f * B;

    const long long* offs = offsets + (size_t)f * (B + 1) + b;
    const long long start = offs[0];
    const long long end   = offs[1];
    const int len = (int)(end - start);

    const long long* vals = values + (size_t)f * T + start;
    const float* tab      = tables + (size_t)f * NROWS * D;
    const int lo          = lane * 4;

    float4v acc = 0.0f;
    float* slots = &stage[wv][0][0];

    if (len >= PIPE) {
        // ---- prologue: fill the ring (PIPE rows in flight) ----
#pragma unroll
        for (int j = 0; j < PIPE; ++j) {
            const long long idx = vals[j];
            ASYNC_LOAD_B128(tab + (size_t)idx * D + lo, &slots[j * D + lo]);
        }
        // ---- steady state: constant-immediate waits, in-order ring ----
        const int steady = len - PIPE;
        for (int i = 0; i < steady; ++i) {
            WAIT_ASYNC(7);                       // PIPE-1: oldest row landed
            const int s = i & (PIPE - 1);
            acc += *(const float4v*)&slots[(size_t)s * D + lo];
            WAIT_DS0();                          // slot drained before DMA reuse
            const long long idx = vals[i + PIPE];
            ASYNC_LOAD_B128(tab + (size_t)idx * D + lo, &slots[s * D + lo]);
        }
        // ---- drain ----
        WAIT_ASYNC(0);
#pragma unroll
        for (int k = 0; k < PIPE; ++k) {
            const int s = (steady + k) & (PIPE - 1);
            acc += *(const float4v*)&slots[(size_t)s * D + lo];
        }
    } else if (len > 0) {
        // short bags: direct register gather, two b128 loads in flight
        int i = 0;
        for (; i + 2 <= len; i += 2) {
            const long long i0 = vals[i];
            const long long i1 = vals[i + 1];
            const float4v v0 = *(const float4v*)(tab + (size_t)i0 * D + lo);
            const float4v v1 = *(const float4v*)(tab + (size_t)i1 * D + lo);
            acc += v0;
            acc += v1;
        }
        if (i < len) {
            const long long i0 = vals[i];
            acc += *(const float4v*)(tab + (size_t)i0 * D + lo);
        }
    }
    // len == 0 -> write zeros (output buffer is poisoned by harness)

    float* op = out + ((size_t)b * F + f) * D + lo;
    *(float4v*)op = acc;   // global_store_b128
}

extern "C" void kernel_launch(void* const* d_in, const int* in_sizes, int n_in,
                              void* d_out, int out_size, void* d_ws, size_t ws_size,
                              hipStream_t stream) {
    (void)in_sizes; (void)n_in; (void)out_size; (void)d_ws; (void)ws_size;
    const float*     tables  = (const float*)d_in[0];      // [F, N, D] f32
    const long long* values  = (const long long*)d_in[1];  // [F, T] i64
    const long long* offsets = (const long long*)d_in[2];  // [F, B+1] i64
    float* out = (float*)d_out;                            // [B, F, D] f32

    const int total_bags = F * B;                          // 106496
    dim3 grid(total_bags / WAVES_PER_BLOCK);               // 13312 blocks
    dim3 block(WAVES_PER_BLOCK * 32);                      // 8 wave32s
    embag_pool_kernel<<<grid, block, 0, stream>>>(tables, values, offsets, out);
}